// HiearchicalFFTShiftModel_51024211476627
// MI455X (gfx1250) — compile-verified
//
#include <hip/hip_runtime.h>
#include <math.h>

// ---------------------------------------------------------------------------
// HiearchicalFFTShiftModel on MI455X (gfx1250)
//
// Reference = 25 stages of (2x zero-stuff -> rfft -> phase ramp -> irfft).
// Collapsed analytically into:
//   1) hfs_setup  : per-stage phase fractions f[i] = frac(theta_i/2pi) (f64),
//                   plus stage-0 spectrum S = [1/sqrt2, cos(theta_0)/sqrt2].
//   2) hfs_stage  : 24 in-place spectrum-doubling stages (mirror/conj/rotate,
//                   Im=0 at DC & Nyquist).  Total work O(N), ~33M sincos.
//   3) hfs_pack   : C2R pack z[j] = E + i*O with 1/sqrt(2^25) ortho scale.
//   4) inverse C2C FFT of length 2^24, mixed-radix Stockham 256*256*16*16:
//        2x hfs_fft_pass256 (async global->LDS tile copy, 64 WMMA/wave)
//        2x hfs_fft_pass16  (direct, 16 WMMA/wave)
//   Final complex array w[m] in d_out IS the interleaved real output.
// All butterflies use V_WMMA_F32_16X16X4_F32 (DFT16 matrix as A operand).
// ---------------------------------------------------------------------------

typedef __attribute__((ext_vector_type(2))) float v2f;
typedef __attribute__((ext_vector_type(8))) float v8f;
typedef __attribute__((ext_vector_type(2))) int   v2i;

#define TWO_PI_F 6.28318530717958647692f
#define RSQRT2_F 0.70710678118654752440f
#define M_HALF   (1u << 24)   // complex IFFT length
#define N_STAGES 25

// ---------------------------------------------------------------------------
// Kernel 0: phase fractions f[i] = frac(theta_i/2pi) (double), and stage-0 S.
// theta_i = 2*pi * (1 - e_i/2^i) * 2^i/(2^i+1)
// ---------------------------------------------------------------------------
__global__ void hfs_setup(const float* __restrict__ elements,
                          double* __restrict__ f,
                          float2* __restrict__ S) {
    int i = threadIdx.x;
    if (i < N_STAGES) {
        double s  = (double)elements[i] * ldexp(1.0, -i);
        double p2 = (double)(1u << i);
        double v  = (1.0 - s) * (p2 / (p2 + 1.0));
        v = v - floor(v);          // frac, v was in (0,2)
        f[i] = v;
        if (i == 0) {              // S_0 = [1/sqrt2, cos(theta_0)/sqrt2]
            S[0] = make_float2(RSQRT2_F, 0.0f);
            S[1] = make_float2(__cosf(TWO_PI_F * (float)v) * RSQRT2_F, 0.0f);
        }
    }
}

// ---------------------------------------------------------------------------
// Kernel 1: one spectrum-doubling stage, fully in place.
// S_i[k]      = S_{i-1}[k]      /sqrt2 * e^{+ik*theta_i}           k <= 2^{i-1}
// S_i[2^i-k'] = conj(S_{i-1}[k'])/sqrt2 * e^{+i(2^i-k')theta_i}    0<k'<2^{i-1}
// S_i[2^i]    = S_{i-1}[0]      /sqrt2 * e^{+i*2^i*theta_i}, Im->0
// Thread k' reads S[k'] and writes {k', 2^i-k'}: race-free in place.
// ---------------------------------------------------------------------------
__global__ void __launch_bounds__(256)
hfs_stage(float2* __restrict__ S, const double* __restrict__ f, int i) {
    unsigned kp   = blockIdx.x * blockDim.x + threadIdx.x;
    unsigned half = 1u << (i - 1);
    unsigned full = 1u << i;
    if (kp > half) return;
    float2 v  = S[kp];
    double fi = f[i];

    // low output k = kp
    double fr = (double)kp * fi; fr -= floor(fr);
    float s, c; __sincosf(TWO_PI_F * (float)fr, &s, &c);
    float2 lo = make_float2((v.x * c - v.y * s) * RSQRT2_F,
                            (v.x * s + v.y * c) * RSQRT2_F);
    if (kp == 0u) lo.y = 0.0f;                     // DC stays real

    // high output k2 = 2^i - kp (skip duplicate when kp == half)
    if (kp < half) {
        unsigned k2 = full - kp;
        float vy = (kp == 0u) ? v.y : -v.y;        // conj for reflected bins
        double fr2 = (double)k2 * fi; fr2 -= floor(fr2);
        float s2, c2; __sincosf(TWO_PI_F * (float)fr2, &s2, &c2);
        float2 hv = make_float2((v.x * c2 - vy * s2) * RSQRT2_F,
                                (v.x * s2 + vy * c2) * RSQRT2_F);
        if (kp == 0u) hv.y = 0.0f;                 // Nyquist Im dropped
        S[k2] = hv;
    }
    S[kp] = lo;
}

// ---------------------------------------------------------------------------
// Kernel 2: C2R pack.  z[j] = E + i*O,
//   E = (S[j]+conj(S[M-j]))/2,  O = (S[j]-conj(S[M-j]))/2 * e^{+i*pi*j/M}
// scaled by 1/sqrt(2^25).
// ---------------------------------------------------------------------------
__global__ void __launch_bounds__(256)
hfs_pack(const float2* __restrict__ S, float2* __restrict__ z) {
    unsigned j = blockIdx.x * blockDim.x + threadIdx.x;   // 0 .. 2^24-1
    float2 Sj = S[j];
    float2 Sm = S[M_HALF - j];

    float Ex = 0.5f * (Sj.x + Sm.x);
    float Ey = 0.5f * (Sj.y - Sm.y);
    float Ox = 0.5f * (Sj.x - Sm.x);
    float Oy = 0.5f * (Sj.y + Sm.y);

    float s, c;   // fraction j/2^25 is exact in f32
    __sincosf(TWO_PI_F * ((float)j * (1.0f / 33554432.0f)), &s, &c);
    float Orx = Ox * c - Oy * s;
    float Ory = Ox * s + Oy * c;

    const float SC = 0.00017263349150062288f;  // 2^-12.5 = 1/sqrt(2^25)
    z[j] = make_float2((Ex - Ory) * SC, (Ey + Orx) * SC);
}

// ---------------------------------------------------------------------------
// DFT16 matrix chunks for the A operand: F[m][k] = exp(+2*pi*i*m*k/16).
// 16x4 f32 A layout: VGPR e holds K = 4q + 2*hi + e for row m = lane%16.
// ---------------------------------------------------------------------------
__device__ __forceinline__ void dft16_mats(unsigned m, unsigned hi,
                                           v2f Fr[4], v2f Fi[4], v2f Fm[4]) {
#pragma unroll
    for (int q = 0; q < 4; ++q) {
        unsigned k0 = 4u * (unsigned)q + 2u * hi;
#pragma unroll
        for (int e = 0; e < 2; ++e) {
            float a = TWO_PI_F * (float)((m * (k0 + (unsigned)e)) & 15u) * 0.0625f;
            float s, c;
            __sincosf(a, &s, &c);
            Fr[q][e] = c; Fi[q][e] = s; Fm[q][e] = -s;
        }
    }
}

#define CWMMA(Re, Im, Ar, Ai, An, Br, Bi)                                      \
    do {                                                                       \
        _Pragma("unroll")                                                      \
        for (int q_ = 0; q_ < 4; ++q_)                                         \
            Re = __builtin_amdgcn_wmma_f32_16x16x4_f32(false, Ar[q_], false,   \
                     Br[q_], (short)0, Re, false, false);                      \
        _Pragma("unroll")                                                      \
        for (int q_ = 0; q_ < 4; ++q_)                                         \
            Re = __builtin_amdgcn_wmma_f32_16x16x4_f32(false, An[q_], false,   \
                     Bi[q_], (short)0, Re, false, false);                      \
        _Pragma("unroll")                                                      \
        for (int q_ = 0; q_ < 4; ++q_)                                         \
            Im = __builtin_amdgcn_wmma_f32_16x16x4_f32(false, Ai[q_], false,   \
                     Br[q_], (short)0, Im, false, false);                      \
        _Pragma("unroll")                                                      \
        for (int q_ = 0; q_ < 4; ++q_)                                         \
            Im = __builtin_amdgcn_wmma_f32_16x16x4_f32(false, Ar[q_], false,   \
                     Bi[q_], (short)0, Im, false, false);                      \
    } while (0)

// async-copy pointer casts: builtin wants int2* in AS1 (global) / AS3 (LDS)
#define GASY(p) ((__attribute__((address_space(1))) v2i*)(v2i*)(float2*)(p))
#define LASY(p) ((__attribute__((address_space(3))) v2i*)(v2i*)(p))

// ---------------------------------------------------------------------------
// Kernel 3: Stockham radix-256 pass (logNs = 0 or 8) through LDS.
// Each workgroup: 16 transforms (j = 16*blk .. +15), 256x16 complex tile.
//   async global->LDS copy (overlapped with DFT16 constant gen)
//   -> stage A (outer twiddle + DFT16 over r2, W256^{k1*r1})
//   -> abuf -> stage B (DFT16 over r1) -> xbuf -> coalesced store.
// Padded LDS strides (17 / 273) keep both axes bank-conflict-free.
// ---------------------------------------------------------------------------
__global__ void __launch_bounds__(256)
hfs_fft_pass256(const float2* __restrict__ in, float2* __restrict__ out,
                int logNs) {
    extern __shared__ float2 smem[];
    float2* xbuf = smem;           // n*17 + jj           (max 4350)
    float2* abuf = smem + 4352;    // jj*273 + k1 + 17*r1 (max 4367)

    const unsigned tid   = threadIdx.x;
    const unsigned lane  = tid & 31u;
    const unsigned wv    = tid >> 5;       // wave in block: 0..7
    const unsigned hi    = lane >> 4;
    const unsigned lm    = lane & 15u;
    const unsigned jbase = blockIdx.x << 4;

    const unsigned Ns = 1u << logNs;
    const float invDen = ldexpf(1.0f, -(logNs + 8));   // 1/(256*Ns), exact

    // ---- async DMA of the 256x16 tile into LDS (ASYNCcnt-tracked)
#pragma unroll 4
    for (int it = 0; it < 16; ++it) {
        unsigned e  = (unsigned)it * 256u + tid;
        unsigned n  = e >> 4;              // 0..255
        unsigned jj = e & 15u;
        unsigned j  = jbase + jj;
        __builtin_amdgcn_global_load_async_to_lds_b64(
            GASY(in + (j + (n << 16))),            // stride M/256 = 65536
            LASY(xbuf + (n * 17u + jj)), 0, 0);
    }

    // overlap: build DFT16 A-operand while the DMA runs
    v2f Fr[4], Fi[4], Fm[4];
    dft16_mats(lm, hi, Fr, Fi, Fm);

    __builtin_amdgcn_s_wait_asynccnt(0);
    __syncthreads();

    // ---- stage A: outer twiddle W^{+n*(j mod Ns)/(256*Ns)} folded into read,
    //      then A[k1,r1] = sum_{r2} x[r1+16r2] W16^{r2k1}, * W256^{k1*r1}
#pragma unroll
    for (int tt = 0; tt < 2; ++tt) {
        unsigned jj = 2u * wv + (unsigned)tt;
        unsigned jm = (jbase + jj) & (Ns - 1u);
        v2f Br[4], Bi[4];
#pragma unroll
        for (int q = 0; q < 4; ++q)
#pragma unroll
            for (int e = 0; e < 2; ++e) {
                unsigned r2 = 4u * (unsigned)q + 2u * hi + (unsigned)e;
                unsigned n  = lm + 16u * r2;
                float2 v = xbuf[n * 17u + jj];
                float s, c;   // numerator n*jm < 256*Ns <= 2^24: exact in f32
                __sincosf(TWO_PI_F * ((float)(n * jm) * invDen), &s, &c);
                Br[q][e] = v.x * c - v.y * s;
                Bi[q][e] = v.x * s + v.y * c;
            }
        v8f Re = {0.f,0.f,0.f,0.f,0.f,0.f,0.f,0.f};
        v8f Im = Re;
        CWMMA(Re, Im, Fr, Fi, Fm, Br, Bi);
#pragma unroll
        for (int d = 0; d < 8; ++d) {      // twiddle W256^{+k1*r1}, write abuf
            unsigned k1 = (unsigned)d + 8u * hi;
            float s, c;
            __sincosf(TWO_PI_F * ((float)(k1 * lm) * 0.00390625f), &s, &c);
            abuf[jj * 273u + k1 + 17u * lm] =
                make_float2(Re[d] * c - Im[d] * s, Re[d] * s + Im[d] * c);
        }
    }
    __syncthreads();

    // ---- stage B: u[k1+16k2] = sum_{r1} A[k1,r1] W16^{r1k2}  -> xbuf[k*17+jj]
#pragma unroll
    for (int tt = 0; tt < 2; ++tt) {
        unsigned jj = 2u * wv + (unsigned)tt;
        v2f Br[4], Bi[4];
#pragma unroll
        for (int q = 0; q < 4; ++q)
#pragma unroll
            for (int e = 0; e < 2; ++e) {
                unsigned r1 = 4u * (unsigned)q + 2u * hi + (unsigned)e;
                float2 v = abuf[jj * 273u + lm + 17u * r1];
                Br[q][e] = v.x; Bi[q][e] = v.y;
            }
        v8f Re = {0.f,0.f,0.f,0.f,0.f,0.f,0.f,0.f};
        v8f Im = Re;
        CWMMA(Re, Im, Fr, Fi, Fm, Br, Bi);
#pragma unroll
        for (int d = 0; d < 8; ++d) {
            unsigned k = lm + 16u * ((unsigned)d + 8u * hi);
            xbuf[k * 17u + jj] = make_float2(Re[d], Im[d]);
        }
    }
    __syncthreads();

    // ---- store: out[(j div Ns)*256Ns + (j mod Ns) + k*Ns], coalesced
    if (logNs == 0) {
#pragma unroll 4
        for (int it = 0; it < 16; ++it) {  // k = tid, jj = it: contiguous
            unsigned jj = (unsigned)it;
            out[(jbase + jj) * 256u + tid] = xbuf[tid * 17u + jj];
        }
    } else {
#pragma unroll 4
        for (int it = 0; it < 16; ++it) {
            unsigned e  = (unsigned)it * 256u + tid;
            unsigned k  = e >> 4;
            unsigned jj = e & 15u;
            unsigned j  = jbase + jj;
            unsigned jm = j & (Ns - 1u);
            out[(j >> logNs) * (Ns << 8) + jm + k * Ns] = xbuf[k * 17u + jj];
        }
    }
}

// ---------------------------------------------------------------------------
// Kernel 4: direct Stockham radix-16 pass (t = 4, 5; Ns = 16^t).
// One wave = 16 transforms; A = DFT16, B = twiddled data; 16 WMMA.
// ---------------------------------------------------------------------------
__global__ void __launch_bounds__(256)
hfs_fft_pass16(const float2* __restrict__ in, float2* __restrict__ out, int t) {
    const unsigned lane = threadIdx.x & 31u;
    const unsigned wave = blockIdx.x * (blockDim.x >> 5) + (threadIdx.x >> 5);
    const unsigned hi   = lane >> 4;
    const unsigned m    = lane & 15u;
    const unsigned j    = (wave << 4) + m;

    v2f Fr[4], Fi[4], Fm[4];
    dft16_mats(m, hi, Fr, Fi, Fm);

    const unsigned Ns = 1u << (4 * t);
    const unsigned jm = j & (Ns - 1u);
    const float invDen = ldexpf(1.0f, -4 * (t + 1));   // 1/(16*Ns), exact

    v2f Br[4], Bi[4];
#pragma unroll
    for (int q = 0; q < 4; ++q)
#pragma unroll
        for (int e = 0; e < 2; ++e) {
            unsigned r = 4u * (unsigned)q + 2u * hi + (unsigned)e;
            float2 v = in[j + (r << 20)];              // stride M/16 = 2^20
            float s, c;
            __sincosf(TWO_PI_F * ((float)(r * jm) * invDen), &s, &c);
            Br[q][e] = v.x * c - v.y * s;
            Bi[q][e] = v.x * s + v.y * c;
        }

    v8f Re = {0.f,0.f,0.f,0.f,0.f,0.f,0.f,0.f};
    v8f Im = Re;
    CWMMA(Re, Im, Fr, Fi, Fm, Br, Bi);

    const unsigned base = (j >> (4 * t)) * (Ns << 4) + jm;
#pragma unroll
    for (int d = 0; d < 8; ++d) {
        unsigned k = (unsigned)d + 8u * hi;
        out[base + k * Ns] = make_float2(Re[d], Im[d]);
    }
}

// ---------------------------------------------------------------------------
// Host launcher.  d_in[0] = elements (25 f32).  d_out = 2^25 f32.
// d_ws: doubles f[25] at 0; S / FFT-pong buffer (2^24+1 float2) at +1024.
// pack -> d_out; 4 passes ping-pong d_out <-> ws, ending in d_out.
// ---------------------------------------------------------------------------
extern "C" void kernel_launch(void* const* d_in, const int* in_sizes, int n_in,
                              void* d_out, int out_size, void* d_ws, size_t ws_size,
                              hipStream_t stream) {
    (void)in_sizes; (void)n_in; (void)out_size; (void)ws_size;

    const float* elements = (const float*)d_in[0];
    double* f   = (double*)d_ws;
    float2* SW  = (float2*)((char*)d_ws + 1024);   // S, then FFT pong buffer
    float2* OUT = (float2*)d_out;

    hfs_setup<<<1, 32, 0, stream>>>(elements, f, SW);
    for (int i = 1; i <= 24; ++i) {
        unsigned n = (1u << (i - 1)) + 1u;
        hfs_stage<<<(n + 255u) / 256u, 256, 0, stream>>>(SW, f, i);
    }
    hfs_pack<<<(1u << 24) / 256, 256, 0, stream>>>(SW, OUT);

    const size_t lds = (4352u + 4368u) * sizeof(float2);   // ~69.8 KB
    hfs_fft_pass256<<<4096, 256, lds, stream>>>(OUT, SW, 0);   // Ns=1
    hfs_fft_pass256<<<4096, 256, lds, stream>>>(SW, OUT, 8);   // Ns=256
    hfs_fft_pass16 <<<131072, 256, 0, stream>>>(OUT, SW, 4);   // Ns=16^4
    hfs_fft_pass16 <<<131072, 256, 0, stream>>>(SW, OUT, 5);   // Ns=16^5
}